// LastMessageAggregator_8881992368208
// MI455X (gfx1250) — compile-verified
//
#include <hip/hip_runtime.h>
#include <hip/hip_bf16.h>

// Pure data-movement workload: segment argmax (scatter) + row gather.
// ~1.0 GB of HBM traffic, 0 FLOPs -> bandwidth bound (~45us floor @ 23.3 TB/s).
// gfx1250 paths used: wave32 b128 row copies, non-temporal TH hints,
// global_prefetch_b8 software pipeline, L2-resident atomicMax table.

typedef float v4f __attribute__((ext_vector_type(4)));

// ---------------- Phase 1: init last-event table to -1 (every call) --------
__global__ void lm_init_last(int* __restrict__ last, int N) {
  int i = blockIdx.x * blockDim.x + threadIdx.x;
  if (i < N) last[i] = -1;
}

// ---------------- Phase 2: scatter-max of event index per node -------------
__global__ void lm_scatter_max(const int* __restrict__ node_ids,
                               int* __restrict__ last, int E) {
  int i = blockIdx.x * blockDim.x + threadIdx.x;
  if (i < E) {
    // 4 MB table lives in L2; atomics resolve there.
    atomicMax(&last[node_ids[i]], i);
  }
}

// ---------------- Phase 3: one wave per node, gather row + time ------------
// D floats per row; lane copies one float4 per 128-column chunk
// (D=128 -> exactly one global_load_b128 + global_store_b128 per lane).
__global__ void lm_gather(const float* __restrict__ msg,
                          const float* __restrict__ times,
                          const int* __restrict__ last,
                          float* __restrict__ out_valid,
                          float* __restrict__ out_msg,
                          float* __restrict__ out_times,
                          int N, int D) {
  const int lane = threadIdx.x & 31;
  const int wave = (blockIdx.x * blockDim.x + threadIdx.x) >> 5;
  const int nwav = (gridDim.x * blockDim.x) >> 5;

  int n = wave;
  int l = (n < N) ? last[n] : -1;
  // Prime the pipeline: get this row's cachelines moving (global_prefetch_b8).
  if (l >= 0) __builtin_prefetch(msg + (size_t)l * D + lane * 4, 0, 0);

  while (n < N) {
    // 1-ahead software pipeline: fetch next node's winner and prefetch its row
    // while we stream-copy the current one.
    const int n2 = n + nwav;
    const int l2 = (n2 < N) ? last[n2] : -1;
    if (l2 >= 0) __builtin_prefetch(msg + (size_t)l2 * D + lane * 4, 0, 0);

    float* __restrict__ orow = out_msg + (size_t)n * D;
    if (l >= 0) {
      const float* __restrict__ irow = msg + (size_t)l * D;
      if (lane == 0) {
        out_valid[n] = 1.0f;
        out_times[n] = times[l];
      }
      // Rows are read once and the 520 MB output exceeds L2 -> non-temporal
      // on both sides to keep L2 for the atomic table / last_idx.
      for (int c = lane * 4; c < D; c += 128) {
        v4f v = __builtin_nontemporal_load((const v4f*)(irow + c));
        __builtin_nontemporal_store(v, (v4f*)(orow + c));
      }
    } else {
      if (lane == 0) {
        out_valid[n] = 0.0f;
        out_times[n] = 0.0f;
      }
      v4f z = (v4f)(0.0f);
      for (int c = lane * 4; c < D; c += 128)
        __builtin_nontemporal_store(z, (v4f*)(orow + c));
    }

    n = n2;
    l = l2;
  }
}

extern "C" void kernel_launch(void* const* d_in, const int* in_sizes, int n_in,
                              void* d_out, int out_size, void* d_ws, size_t ws_size,
                              hipStream_t stream) {
  const int*   node_ids = (const int*)d_in[0];
  const float* msg      = (const float*)d_in[1];
  const float* times    = (const float*)d_in[2];
  // d_in[3] (num_nodes) is device-resident; derive sizes on host instead
  // (graph-capture safe, no sync memcpy):
  const int E = in_sizes[0];
  const int D = in_sizes[1] / in_sizes[0];      // 128
  const int N = out_size / (D + 2);             // valid[N] + msg[N*D] + times[N]

  int* last = (int*)d_ws;                       // needs N*4 bytes (4 MB)

  float* out_valid = (float*)d_out;             // [N]
  float* out_msg   = out_valid + N;             // [N, D]
  float* out_times = out_msg + (size_t)N * D;   // [N]

  // Phase 1: reset workspace every call (harness does not re-poison).
  lm_init_last<<<(N + 255) / 256, 256, 0, stream>>>(last, N);

  // Phase 2: scatter argmax of event index.
  lm_scatter_max<<<(E + 255) / 256, 256, 0, stream>>>(node_ids, last, E);

  // Phase 3: one wave per node; cap grid so each wave loops a few nodes,
  // giving the 1-ahead prefetch pipeline real work.
  int blocks = (N + 7) / 8;                     // 8 waves per 256-thread block
  if (blocks > 16384) blocks = 16384;           // ~131K waves, ~8 nodes each
  lm_gather<<<blocks, 256, 0, stream>>>(msg, times, last,
                                        out_valid, out_msg, out_times, N, D);
}